// dEMD_82274393523065
// MI455X (gfx1250) — compile-verified
//
#include <hip/hip_runtime.h>
#include <stdint.h>

// Greedy multi-marginal EMD (d=8, n=1024) — inherently sequential scalar loop.
// MI455X strategy: one wave32, TDM (tensor_load_to_lds) stages the 32KB input
// into LDS in a single DMA, then a register-resident select-tree loop with a
// one-iteration LDS prefetch for the advancing marginal's next front value.

#define EMD_D 8
#define EMD_N 1024

typedef unsigned int v4u __attribute__((ext_vector_type(4)));
typedef int          v8i __attribute__((ext_vector_type(8)));
typedef int          v4i __attribute__((ext_vector_type(4)));

__device__ __forceinline__ int imin(int a, int b) { return a < b ? a : b; }
__device__ __forceinline__ int imax(int a, int b) { return a > b ? a : b; }

__global__ __launch_bounds__(32) void demd_greedy_kernel(const float* __restrict__ x,
                                                         float* __restrict__ out) {
    __shared__ float lds[EMD_D * EMD_N];   // 32 KB of 320 KB WGP LDS

    // ---------------- Stage x (8x1024 f32) into LDS ----------------
#if defined(__has_builtin) && __has_builtin(__builtin_amdgcn_tensor_load_to_lds) && \
    __has_builtin(__builtin_amdgcn_s_wait_tensorcnt)
    {
        // Tensor DMA Descriptor (D#) for a 1-D tile of 8192 f32 elements.
        // Group 0 (128b): [1:0]=count=1, [63:32]=lds_addr, [120:64]=global_addr,
        //                 [127:126]=type=2.
        uint64_t gaddr = (uint64_t)(uintptr_t)x;
        // Flat LDS-aperture address: addr[31:0] is the LDS byte offset (ISA 10.2).
        uint32_t laddr = (uint32_t)(uintptr_t)lds;
        v4u g0;
        g0.x = 1u;                                   // count=1 (valid user descriptor)
        g0.y = laddr;                                // lds_addr (bytes)
        g0.z = (uint32_t)gaddr;                      // global_addr[31:0]
        g0.w = (uint32_t)((gaddr >> 32) & 0x01FFFFFFu) | 0x80000000u; // addr[56:32]|type=2

        // Group 1 (256b): workgroup_mask=0, data_size=2 (4B), no barrier/iterate/pad,
        // tensor_dim0 = 8192 (bits 79:48), tensor_dim1 = 1 (bits 111:80),
        // tile_dim0 = 8192 (bits 127:112), tile_dim1/2 = 0, dim0_stride = 8192.
        v8i g1;
        g1[0] = 0x00020000;                          // data_size=4B at [17:16]
        g1[1] = (int)(((EMD_D * EMD_N) & 0xFFFF) << 16);        // tensor_dim0 lo16 @ [63:48]
        g1[2] = (int)(((EMD_D * EMD_N) >> 16) & 0xFFFF) | (1 << 16); // dim0 hi16 | tensor_dim1=1
        g1[3] = (int)(((EMD_D * EMD_N) & 0xFFFF) << 16);        // tile_dim0 @ [127:112]
        g1[4] = 0;                                   // tile_dim1=0, tile_dim2=0
        g1[5] = EMD_D * EMD_N;                       // tensor_dim0_stride lo32
        g1[6] = 0;
        g1[7] = 0;
        v4i z4 = {0, 0, 0, 0};                       // groups 2/3 unused (<=2D tensor)
        v8i z8 = {0, 0, 0, 0, 0, 0, 0, 0};           // extra group (clang-23 6-arg form)

        // 6-arg amdgpu-toolchain signature: (v4u, v8i, v4i, v4i, v8i, i32 cpol)
        __builtin_amdgcn_tensor_load_to_lds(g0, g1, z4, z4, z8, 0);
        __builtin_amdgcn_s_wait_tensorcnt(0);        // TDM done -> LDS valid for this wave
        // Compiler barrier: the builtin only saw the LDS address as an integer,
        // so force it to treat LDS as written before the reads below.
        asm volatile("" : : "r"(lds) : "memory");
    }
#else
    // Fallback: cooperative copy (single wave, 64 float4 per lane).
    for (int t = threadIdx.x; t < (EMD_D * EMD_N) / 4; t += blockDim.x)
        ((float4*)lds)[t] = ((const float4*)x)[t];
    __syncthreads();
#endif

    // ---------------- Sequential greedy loop (all lanes run it redundantly) ----------
    float r[EMD_D];    // current front remaining mass per marginal
    float nx[EMD_D];   // prefetched next front value per marginal
    int   id[EMD_D];   // front pointer per marginal
#pragma unroll
    for (int j = 0; j < EMD_D; ++j) {
        r[j]  = lds[j * EMD_N + 0];
        nx[j] = lds[j * EMD_N + 1];
        id[j] = 0;
    }

    float obj = 0.0f;
    int   mx  = 0;     // max(idx) — pointers only grow, maintained incrementally

    for (;;) {
        // First-min argmin tree over r[0..7]; strict '<' keeps the lower index on
        // ties (matches jnp.argmin). Also carries id[ind] (sel) through the tree.
        float v01, v23, v45, v67, v03, v47, mv;
        int   j01, j23, j45, j67, j03, j47, ind;
        int   s01, s23, s45, s67, s03, s47, sel;
        { bool c = r[1] < r[0]; v01 = c ? r[1] : r[0]; j01 = c ? 1 : 0; s01 = c ? id[1] : id[0]; }
        { bool c = r[3] < r[2]; v23 = c ? r[3] : r[2]; j23 = c ? 3 : 2; s23 = c ? id[3] : id[2]; }
        { bool c = r[5] < r[4]; v45 = c ? r[5] : r[4]; j45 = c ? 5 : 4; s45 = c ? id[5] : id[4]; }
        { bool c = r[7] < r[6]; v67 = c ? r[7] : r[6]; j67 = c ? 7 : 6; s67 = c ? id[7] : id[6]; }
        { bool c = v23 < v01; v03 = c ? v23 : v01; j03 = c ? j23 : j01; s03 = c ? s23 : s01; }
        { bool c = v67 < v45; v47 = c ? v67 : v45; j47 = c ? j67 : j45; s47 = c ? s67 : s45; }
        { bool c = v47 < v03; mv  = c ? v47 : v03; ind = c ? j47 : j03; sel = c ? s47 : s03; }

        // cost = max(idx) - min(idx) BEFORE advancing (independent of the fp tree).
        int mn = imin(imin(imin(id[0], id[1]), imin(id[2], id[3])),
                      imin(imin(id[4], id[5]), imin(id[6], id[7])));
        // Separate mul then add (no FMA) to match the reference's rounding.
        obj = __fadd_rn(obj, __fmul_rn((float)(mx - mn), mv));

        // Subtract minval from every front; the argmin row takes its prefetched
        // fresh value (its old front is exactly mv -> 0, then pointer advances).
        int newidx = sel + 1;
#pragma unroll
        for (int j = 0; j < EMD_D; ++j) {
            bool e = (j == ind);
            r[j]  = e ? nx[j] : (r[j] - mv);
            id[j] = e ? newidx : id[j];
        }
        mx = imax(mx, newidx);

        if (newidx == EMD_N) break;   // a pointer hit n: remaining steps are no-ops

        // Prefetch the advanced marginal's next-next value (clamped address; the
        // value is never consumed when the clamp engages). This LDS load's latency
        // is hidden behind the next iteration's select tree.
        int a2 = newidx + 1;
        if (a2 > EMD_N - 1) a2 = EMD_N - 1;
        float ld = lds[ind * EMD_N + a2];
#pragma unroll
        for (int j = 0; j < EMD_D; ++j)
            nx[j] = (j == ind) ? ld : nx[j];
    }

    if (threadIdx.x == 0) out[0] = obj;
}

extern "C" void kernel_launch(void* const* d_in, const int* in_sizes, int n_in,
                              void* d_out, int out_size, void* d_ws, size_t ws_size,
                              hipStream_t stream) {
    (void)in_sizes; (void)n_in; (void)d_ws; (void)ws_size; (void)out_size;
    const float* x = (const float*)d_in[0];   // (8, 1024) float32, row-major
    float* out = (float*)d_out;               // scalar float32
    // One wave32: the algorithm is a single sequential dependence chain.
    demd_greedy_kernel<<<1, 32, 0, stream>>>(x, out);
}